// HMMLayer_81664508166610
// MI455X (gfx1250) — compile-verified
//
#include <hip/hip_runtime.h>
#include <math.h>

// Problem constants
#define BB   64
#define TT   4096
#define IN   256
#define HH   128
#define KK   4
#define NROWS (BB * TT)        // 262144
#define SEG   64               // segment length for the scan
#define NSEG  (TT / SEG)       // 64 segments
#define HPAD  132              // padded H row stride in LDS (conflict-free)

typedef float v2f __attribute__((ext_vector_type(2)));
typedef float v8f __attribute__((ext_vector_type(8)));

// Branch-free tanh: exact at +-inf, abs err <= ~3e-7 (vs branchy ocml tanhf).
__device__ __forceinline__ float tanh_fast(float x) {
    float e = __expf(2.0f * x);          // inf for large x, 0 for very negative
    return 1.0f - 2.0f / (e + 1.0f);
}

// ---------------------------------------------------------------------------
// Kernel 1: emission net.  log_B = log_softmax(tanh(X@W1+b1)@W2+b2, axis=-1)
// Block: 256 threads = 8 waves, 32 rows per wave -> 256 rows per block.
// GEMM1 runs on V_WMMA_F32_16X16X4_F32 (fp32 matrix pipe, exact vs reference).
//
// W1 is staged in LDS pair-interleaved + XOR-swizzled so each lane's B-frag
// (W1[k][n], W1[k+1][n]) is one aligned ds_load_b64 into an even VGPR pair,
// and lane-halves (parity of pair index == lh) hit disjoint bank halves.
//   phys dword off(k, n) = (k>>1)*256 + (((n<<1) + (k&1)) ^ (((k>>1)&1)<<5))
// ---------------------------------------------------------------------------
__global__ __launch_bounds__(256)
void emission_kernel(const float* __restrict__ X,
                     const float* __restrict__ W1,
                     const float* __restrict__ b1,
                     const float* __restrict__ W2,
                     const float* __restrict__ b2,
                     float* __restrict__ logB) {
    __shared__ __align__(16) float W1s[IN * HH];      // 128 KB (swizzled)
    __shared__ __align__(16) float Hs[8 * 32 * HPAD]; // 132 KB (8 waves x 32x132)
    __shared__ __align__(16) float W2s[HH * KK];      // 2 KB
    __shared__ float b1s[HH];
    __shared__ float b2s[KK];

    const int tid = threadIdx.x;

    // Cooperative stage of W1 (swizzled) / W2 / b1 / b2 into LDS
    {
        const float4* w1v = (const float4*)W1;
        #pragma unroll
        for (int i = 0; i < (IN * HH / 4) / 256; ++i) {
            const int idx4 = i * 256 + tid;
            const float4 v = w1v[idx4];
            const int e = idx4 * 4;          // element index; 4 elems share k
            const int k = e >> 7;            // row of W1 (K index)
            const int n = e & 127;           // col of W1 (N index), n..n+3
            const int p = k >> 1;            // pair index
            const int sw = (p & 1) << 5;     // bank-half swizzle
            const int b0 = k & 1;            // which half of the pair
            float* base = &W1s[p * 256 + b0];
            base[((n + 0) << 1) ^ sw] = v.x;
            base[((n + 1) << 1) ^ sw] = v.y;
            base[((n + 2) << 1) ^ sw] = v.z;
            base[((n + 3) << 1) ^ sw] = v.w;
        }
        W2s[tid] = W2[tid];
        W2s[tid + 256] = W2[tid + 256];
        if (tid < HH) b1s[tid] = b1[tid];
        if (tid < KK) b2s[tid] = b2[tid];
    }
    __syncthreads();

    const int wave = tid >> 5;
    const int lane = tid & 31;
    const int lm = lane & 15;       // row within tile (A) / col within tile (B)
    const int lh = lane >> 4;       // lane half -> K offset of 2*lh

    const int row0 = blockIdx.x * 256 + wave * 32;

    // fp32 WMMA accumulators: two 16-row tiles x 8 column tiles (128 cols)
    v8f acc0[8], acc1[8];
    #pragma unroll
    for (int j = 0; j < 8; ++j) { acc0[j] = (v8f)0.0f; acc1[j] = (v8f)0.0f; }

    const float* xp0 = X + (size_t)(row0 + lm) * IN + 2 * lh;
    const float* xp1 = xp0 + 16 * IN;
    const int swr = lh << 5;        // reader swizzle (pair parity == lh)

    for (int kk = 0; kk < IN; kk += 4) {
        // A-frags (16x4 fp32): v0 holds K=kk (lanes0-15) / kk+2 (lanes16-31),
        // v1 holds kk+1 / kk+3  -> one float2 per lane.
        v2f a0 = *(const v2f*)(xp0 + kk);
        v2f a1 = *(const v2f*)(xp1 + kk);
        const float* w1p = &W1s[((kk >> 1) + lh) * 256];
        #pragma unroll
        for (int j = 0; j < 8; ++j) {
            const int nj = j * 16 + lm;
            // (W1[kk+2lh][nj], W1[kk+2lh+1][nj]) as one aligned b64 load
            const v2f bf = *(const v2f*)&w1p[(nj << 1) ^ swr];
            acc0[j] = __builtin_amdgcn_wmma_f32_16x16x4_f32(
                false, a0, false, bf, (short)0, acc0[j], false, false);
            acc1[j] = __builtin_amdgcn_wmma_f32_16x16x4_f32(
                false, a1, false, bf, (short)0, acc1[j], false, false);
        }
    }

    // bias + tanh, stage H tile to LDS.
    // C layout: vgpr r holds (M = r + 8*lh, N = lm) of each 16x16 tile.
    float* hw = &Hs[wave * 32 * HPAD];
    #pragma unroll
    for (int j = 0; j < 8; ++j) {
        const float bbv = b1s[j * 16 + lm];
        #pragma unroll
        for (int r = 0; r < 8; ++r) {
            const int m = r + 8 * lh;
            hw[m * HPAD + j * 16 + lm]        = tanh_fast(acc0[j][r] + bbv);
            hw[(m + 16) * HPAD + j * 16 + lm] = tanh_fast(acc1[j][r] + bbv);
        }
    }
    __syncthreads();   // cross-lane H reads below

    // GEMM2 (128x4) + log_softmax over K=4.
    // lane handles rows (m, m+16), k pair kp = 2*lh.
    const int m = lm;
    const int kp = 2 * lh;
    float s00 = 0.f, s01 = 0.f, s10 = 0.f, s11 = 0.f;
    #pragma unroll 4
    for (int n = 0; n < HH; n += 2) {
        const v2f h0 = *(const v2f*)&hw[m * HPAD + n];
        const v2f h1 = *(const v2f*)&hw[(m + 16) * HPAD + n];
        const v2f wa = *(const v2f*)&W2s[n * KK + kp];
        const v2f wb = *(const v2f*)&W2s[(n + 1) * KK + kp];
        s00 += h0.x * wa.x + h0.y * wb.x;
        s01 += h0.x * wa.y + h0.y * wb.y;
        s10 += h1.x * wa.x + h1.y * wb.x;
        s11 += h1.x * wa.y + h1.y * wb.y;
    }
    s00 += b2s[kp]; s01 += b2s[kp + 1];
    s10 += b2s[kp]; s11 += b2s[kp + 1];

    // row m
    {
        float mx = fmaxf(s00, s01);
        float M  = fmaxf(mx, __shfl_xor(mx, 16, 32));
        float e  = expf(s00 - M) + expf(s01 - M);
        e += __shfl_xor(e, 16, 32);
        float lse = M + logf(e);
        v2f o; o.x = s00 - lse; o.y = s01 - lse;
        *(v2f*)&logB[(size_t)(row0 + m) * KK + kp] = o;
    }
    // row m+16
    {
        float mx = fmaxf(s10, s11);
        float M  = fmaxf(mx, __shfl_xor(mx, 16, 32));
        float e  = expf(s10 - M) + expf(s11 - M);
        e += __shfl_xor(e, 16, 32);
        float lse = M + logf(e);
        v2f o; o.x = s10 - lse; o.y = s11 - lse;
        *(v2f*)&logB[(size_t)(row0 + m + 16) * KK + kp] = o;
    }
}

// ---------------------------------------------------------------------------
// Forward-scan helpers (log domain, matches reference max-shifted LSE; the
// banded A makes the 2-term LSE exactly equal to the reference 4-term one
// because exp(-1e9 - m) underflows to exactly 0).
// ---------------------------------------------------------------------------
__device__ __forceinline__ float lse2(float a, float b) {
    float m = fmaxf(a, b);
    float d = fminf(a, b) - m;           // 0 when a==b (incl. both -1e30)
    return m + log1pf(expf(d));
}
__device__ __forceinline__ float lse4(float x0, float x1, float x2, float x3) {
    float m = fmaxf(fmaxf(x0, x1), fmaxf(x2, x3));
    return m + logf(expf(x0 - m) + expf(x1 - m) + expf(x2 - m) + expf(x3 - m));
}
struct LogA { float a00, a01, a11, a12, a22, a23, a33; };
__device__ __forceinline__ LogA load_logA(const float* __restrict__ Araw) {
    float r[16];
    #pragma unroll
    for (int i = 0; i < 16; ++i) r[i] = Araw[i];
    float l[4];
    #pragma unroll
    for (int i = 0; i < 4; ++i)
        l[i] = lse4(r[4 * i], r[4 * i + 1], r[4 * i + 2], r[4 * i + 3]);
    LogA A;
    A.a00 = r[0] - l[0];  A.a01 = r[1] - l[0];
    A.a11 = r[5] - l[1];  A.a12 = r[6] - l[1];
    A.a22 = r[10] - l[2]; A.a23 = r[11] - l[2];
    A.a33 = r[15] - l[3];
    return A;
}
__device__ __forceinline__ void load_logpi(const float* __restrict__ praw, float lp[4]) {
    float p0 = praw[0], p1 = praw[1], p2 = praw[2], p3 = praw[3];
    float l = lse4(p0, p1, p2, p3);
    lp[0] = p0 - l; lp[1] = p1 - l; lp[2] = p2 - l; lp[3] = p3 - l;
}

// ---------------------------------------------------------------------------
// Kernel 2 (pass 1): compose per-segment 4x4 operators in log domain.
// One (b, s) chain per lane; 4096 chains total.  Segment 0 composes t=1..63.
// ---------------------------------------------------------------------------
__global__ __launch_bounds__(256)
void seg_matrix_kernel(const float* __restrict__ logB,
                       const float* __restrict__ Araw,
                       float* __restrict__ Mseg) {
    const int g = blockIdx.x * blockDim.x + threadIdx.x;
    if (g >= BB * NSEG) return;
    const int b = g >> 6, s = g & (NSEG - 1);
    const LogA A = load_logA(Araw);
    const float4* lbp = (const float4*)logB + (size_t)b * TT;

    float M[4][4];
    #pragma unroll
    for (int i = 0; i < 4; ++i)
        #pragma unroll
        for (int j = 0; j < 4; ++j) M[i][j] = (i == j) ? 0.0f : -1e30f;

    int t = s * SEG + (s == 0 ? 1 : 0);
    const int tend = s * SEG + SEG;
    for (; t < tend; ++t) {
        const float4 lb = lbp[t];
        #pragma unroll
        for (int i = 0; i < 4; ++i) {
            float n0 = M[i][0] + A.a00 + lb.x;
            float n1 = lb.y + lse2(M[i][0] + A.a01, M[i][1] + A.a11);
            float n2 = lb.z + lse2(M[i][1] + A.a12, M[i][2] + A.a22);
            float n3 = lb.w + lse2(M[i][2] + A.a23, M[i][3] + A.a33);
            M[i][0] = n0; M[i][1] = n1; M[i][2] = n2; M[i][3] = n3;
        }
    }
    float4* out = (float4*)(Mseg + (size_t)g * 16);
    #pragma unroll
    for (int i = 0; i < 4; ++i)
        out[i] = make_float4(M[i][0], M[i][1], M[i][2], M[i][3]);
}

// ---------------------------------------------------------------------------
// Kernel 3: sequential combine across segments (tiny: 64 chains of 63 steps).
// start[b][s] = alpha_{s*64-1} for s = 1..63.
// ---------------------------------------------------------------------------
__global__ __launch_bounds__(64)
void seg_combine_kernel(const float* __restrict__ logB,
                        const float* __restrict__ praw,
                        const float* __restrict__ Mseg,
                        float* __restrict__ start) {
    const int b = threadIdx.x;
    if (b >= BB) return;
    float lp[4]; load_logpi(praw, lp);
    const float4 lb0 = *((const float4*)logB + (size_t)b * TT);
    float a0 = lp[0] + lb0.x, a1 = lp[1] + lb0.y, a2 = lp[2] + lb0.z, a3 = lp[3] + lb0.w;

    for (int s = 0; s <= NSEG - 2; ++s) {
        const float* M = Mseg + (size_t)(b * NSEG + s) * 16;
        float n0 = lse4(a0 + M[0],  a1 + M[4],  a2 + M[8],  a3 + M[12]);
        float n1 = lse4(a0 + M[1],  a1 + M[5],  a2 + M[9],  a3 + M[13]);
        float n2 = lse4(a0 + M[2],  a1 + M[6],  a2 + M[10], a3 + M[14]);
        float n3 = lse4(a0 + M[3],  a1 + M[7],  a2 + M[11], a3 + M[15]);
        a0 = n0; a1 = n1; a2 = n2; a3 = n3;
        *(float4*)&start[(size_t)(b * NSEG + s + 1) * 4] = make_float4(a0, a1, a2, a3);
    }
}

// ---------------------------------------------------------------------------
// Kernel 4 (pass 3): replay each segment, write log_alpha (and log_prob).
// ---------------------------------------------------------------------------
__global__ __launch_bounds__(256)
void alpha_kernel(const float* __restrict__ logB,
                  const float* __restrict__ Araw,
                  const float* __restrict__ praw,
                  const float* __restrict__ start,
                  float* __restrict__ log_alpha,
                  float* __restrict__ log_prob) {
    const int g = blockIdx.x * blockDim.x + threadIdx.x;
    if (g >= BB * NSEG) return;
    const int b = g >> 6, s = g & (NSEG - 1);
    const LogA A = load_logA(Araw);
    const float4* lbp = (const float4*)logB + (size_t)b * TT;
    float4* aout = (float4*)log_alpha + (size_t)b * TT;

    float a0, a1, a2, a3;
    int t0;
    if (s == 0) {
        float lp[4]; load_logpi(praw, lp);
        const float4 lb = lbp[0];
        a0 = lp[0] + lb.x; a1 = lp[1] + lb.y; a2 = lp[2] + lb.z; a3 = lp[3] + lb.w;
        aout[0] = make_float4(a0, a1, a2, a3);
        t0 = 1;
    } else {
        const float4 st = *(const float4*)&start[(size_t)(b * NSEG + s) * 4];
        a0 = st.x; a1 = st.y; a2 = st.z; a3 = st.w;
        t0 = s * SEG;
    }
    const int tend = s * SEG + SEG;
    for (int t = t0; t < tend; ++t) {
        const float4 lb = lbp[t];
        float n0 = a0 + A.a00 + lb.x;
        float n1 = lb.y + lse2(a0 + A.a01, a1 + A.a11);
        float n2 = lb.z + lse2(a1 + A.a12, a2 + A.a22);
        float n3 = lb.w + lse2(a2 + A.a23, a3 + A.a33);
        a0 = n0; a1 = n1; a2 = n2; a3 = n3;
        aout[t] = make_float4(a0, a1, a2, a3);
    }
    if (s == NSEG - 1) log_prob[b] = lse4(a0, a1, a2, a3);
}

// ---------------------------------------------------------------------------
extern "C" void kernel_launch(void* const* d_in, const int* in_sizes, int n_in,
                              void* d_out, int out_size, void* d_ws, size_t ws_size,
                              hipStream_t stream) {
    const float* X    = (const float*)d_in[0];
    const float* W1   = (const float*)d_in[1];
    const float* b1   = (const float*)d_in[2];
    const float* W2   = (const float*)d_in[3];
    const float* b2   = (const float*)d_in[4];
    const float* Araw = (const float*)d_in[5];
    const float* praw = (const float*)d_in[6];

    float* out        = (float*)d_out;
    float* out_lp     = out;                         // (64,)
    float* out_alpha  = out + BB;                    // (64,4096,4)
    float* out_logB   = out + BB + (size_t)BB * TT * KK;  // (64,4096,4)

    float* wsf   = (float*)d_ws;
    float* Mseg  = wsf;                              // 4096*16 floats
    float* start = wsf + (size_t)BB * NSEG * 16;     // 64*64*4 floats

    emission_kernel<<<NROWS / 256, 256, 0, stream>>>(X, W1, b1, W2, b2, out_logB);
    seg_matrix_kernel<<<(BB * NSEG) / 256, 256, 0, stream>>>(out_logB, Araw, Mseg);
    seg_combine_kernel<<<1, 64, 0, stream>>>(out_logB, praw, Mseg, start);
    alpha_kernel<<<(BB * NSEG) / 256, 256, 0, stream>>>(out_logB, Araw, praw, start,
                                                        out_alpha, out_lp);
}